// construct_low_energy_state_14516989461120
// MI455X (gfx1250) — compile-verified
//
#include <hip/hip_runtime.h>

typedef float v2f __attribute__((ext_vector_type(2)));
typedef float v8f __attribute__((ext_vector_type(8)));

#define NQ 24
#define OUT_N (1u << 24)   // 2^24 complex amplitudes; out = [re(2^24), im(2^24)]

// term[k] = prod_i x[i][(k>>i)&1], normalized.
// Factorization: term[k] = Chi(k[23:16]) * Cmid(k[15:8]) * M16(k[7:4]) * L16(k[3:0])
// Tile = 16x16 (M = k[7:4], N = k[3:0]) -> 256 consecutive outputs per tile.
// Complex outer product per tile done as two V_WMMA_F32_16X16X4_F32 (rank-2 real matmuls).
__global__ __launch_bounds__(256) void kron_state_wmma_kernel(
    const float* __restrict__ xre, const float* __restrict__ xim,
    float* __restrict__ out)
{
    __shared__ float s_xre[2 * NQ], s_xim[2 * NQ];
    __shared__ float s_cmid_re[256], s_cmid_im[256];
    __shared__ float s_chi_re[256],  s_chi_im[256];

    const int tid  = threadIdx.x;     // 0..255
    const int lane = tid & 31;
    const int wave = tid >> 5;        // 0..7

    // Stage the tiny 24x2 complex input in LDS (dynamic indexing friendly).
    if (tid < 2 * NQ) {
        s_xre[tid] = xre[tid];
        s_xim[tid] = xim[tid];
    }
    __syncthreads();

    // 1/norm = rsqrt( prod_i (|x_i0|^2 + |x_i1|^2) )   (norm of a kron = prod of norms)
    float nrm2 = 1.0f;
#pragma unroll
    for (int i = 0; i < NQ; ++i) {
        float r0 = s_xre[2*i+0], i0 = s_xim[2*i+0];
        float r1 = s_xre[2*i+1], i1 = s_xim[2*i+1];
        nrm2 *= r0*r0 + i0*i0 + r1*r1 + i1*i1;
    }
    const float invnorm = rsqrtf(nrm2);

    // Build full 256-entry Cmid (qubits 8..15) and Chi (qubits 16..23) tables:
    // one entry per thread, 7 complex muls each.
    {
        const int j = tid;
        int s = j & 1;
        float cr = s_xre[2*8 + s], ci = s_xim[2*8 + s];
#pragma unroll
        for (int b = 1; b < 8; ++b) {
            s = (j >> b) & 1;
            float ar = s_xre[2*(8+b) + s], ai = s_xim[2*(8+b) + s];
            float nr = cr*ar - ci*ai;
            ci = cr*ai + ci*ar;
            cr = nr;
        }
        s_cmid_re[j] = cr; s_cmid_im[j] = ci;

        s = j & 1;
        cr = s_xre[2*16 + s]; ci = s_xim[2*16 + s];
#pragma unroll
        for (int b = 1; b < 8; ++b) {
            s = (j >> b) & 1;
            float ar = s_xre[2*(16+b) + s], ai = s_xim[2*(16+b) + s];
            float nr = cr*ar - ci*ai;
            ci = cr*ai + ci*ar;
            cr = nr;
        }
        s_chi_re[j] = cr; s_chi_im[j] = ci;
    }

    // Per-lane constants: L16 (qubits 0..3, N axis) and M16 (qubits 4..7, M axis).
    // Lanes >= 16 hold zeros -> K=2,3 rows of the WMMA operands contribute nothing.
    float l16r = 0.f, l16i = 0.f, m16r = 0.f, m16i = 0.f;
    if (lane < 16) {
        const int n = lane;
        int s = n & 1;
        l16r = s_xre[2*0 + s]; l16i = s_xim[2*0 + s];
#pragma unroll
        for (int b = 1; b < 4; ++b) {
            s = (n >> b) & 1;
            float ar = s_xre[2*b + s], ai = s_xim[2*b + s];
            float nr = l16r*ar - l16i*ai;
            l16i = l16r*ai + l16i*ar; l16r = nr;
        }
        s = n & 1;
        m16r = s_xre[2*4 + s]; m16i = s_xim[2*4 + s];
#pragma unroll
        for (int b = 1; b < 4; ++b) {
            s = (n >> b) & 1;
            float ar = s_xre[2*(4+b) + s], ai = s_xim[2*(4+b) + s];
            float nr = m16r*ar - m16i*ai;
            m16i = m16r*ai + m16i*ar; m16r = nr;
        }
        m16r *= invnorm;   // fold normalization into the M-side operand
        m16i *= invnorm;
    }
    __syncthreads();

    // B operands (constant for the whole kernel):
    //   Re-output: B[K0] = Re L16, B[K1] = -Im L16
    //   Im-output: B[K0] = Im L16, B[K1] =  Re L16
    v2f bre, bim;
    bre.x = l16r; bre.y = -l16i;
    bim.x = l16i; bim.y =  l16r;

    // Lane's (m,n) offset inside a tile's 256-float region:
    // D layout: VGPR r -> M=r (lanes 0-15) / M=r+8 (lanes 16-31), N = lane&15.
    const int lane_off = (lane & 15) + ((lane >> 4) << 7);   // +128 for M+8

    const unsigned gw = blockIdx.x * 8u + (unsigned)wave;    // global wave id: 0..8191
    for (int it = 0; it < 8; ++it) {
        const unsigned t = gw * 8u + (unsigned)it;           // tile id: 0..65535
        // C = Chi(t[15:8]) * Cmid(t[7:0])   (uniform across the wave)
        const float chr = s_chi_re[t >> 8],   chii = s_chi_im[t >> 8];
        const float cmr = s_cmid_re[t & 255], cmi  = s_cmid_im[t & 255];
        const float c_r = chr*cmr - chii*cmi;
        const float c_i = chr*cmi + chii*cmr;

        // A operand: Ahat[m] = C * M16[m] * invnorm   (lanes>=16 -> 0)
        v2f a;
        a.x = c_r*m16r - c_i*m16i;   // K=0 row: Re
        a.y = c_r*m16i + c_i*m16r;   // K=1 row: Im

        v8f zero = {};
        // D_re[m][n] = Re(Ahat_m) * Re(L16_n) - Im(Ahat_m) * Im(L16_n) = Re(Ahat_m * L16_n)
        v8f dre = __builtin_amdgcn_wmma_f32_16x16x4_f32(
            false, a, false, bre, (short)0, zero, false, false);
        // D_im[m][n] = Re(Ahat_m) * Im(L16_n) + Im(Ahat_m) * Re(L16_n) = Im(Ahat_m * L16_n)
        v8f dim_ = __builtin_amdgcn_wmma_f32_16x16x4_f32(
            false, a, false, bim, (short)0, zero, false, false);

        float* pre = out + ((size_t)t << 8) + lane_off;   // real part
        float* pim = pre + OUT_N;                          // imag part (+2^24 floats)
#pragma unroll
        for (int r = 0; r < 8; ++r) {
            pre[r * 16] = dre[r];
            pim[r * 16] = dim_[r];
        }
    }
}

extern "C" void kernel_launch(void* const* d_in, const int* in_sizes, int n_in,
                              void* d_out, int out_size, void* d_ws, size_t ws_size,
                              hipStream_t stream) {
    (void)in_sizes; (void)n_in; (void)out_size; (void)d_ws; (void)ws_size;
    const float* xre = (const float*)d_in[0];   // [24,2] float32
    const float* xim = (const float*)d_in[1];   // [24,2] float32
    float* out = (float*)d_out;                 // [2, 2^24] float32

    // 65536 tiles of 256 outputs; 8 waves/block, 8 tiles/wave -> 1024 blocks.
    kron_state_wmma_kernel<<<dim3(1024), dim3(256), 0, stream>>>(xre, xim, out);
}